// MorkySelectiveSSM_89137751261984
// MI455X (gfx1250) — compile-verified
//
#include <hip/hip_runtime.h>
#include <hip/hip_bf16.h>

typedef __attribute__((ext_vector_type(2))) float v2f;
typedef __attribute__((ext_vector_type(8))) float v8f;
typedef unsigned int uint32x4 __attribute__((ext_vector_type(4)));
typedef int int32x4 __attribute__((ext_vector_type(4)));
typedef int int32x8 __attribute__((ext_vector_type(8)));

constexpr int kB = 2;
constexpr int kS = 2048;
constexpr int kH = 512;
constexpr int kN = 64;
constexpr int kL = 64;          // chunk length
constexpr int kC = kS / kL;     // 32 chunks per batch
constexpr int kR = kB * kS;     // 4096 rows total

#if defined(__has_builtin)
#  if __has_builtin(__builtin_amdgcn_tensor_load_to_lds) && \
      __has_builtin(__builtin_amdgcn_s_wait_tensorcnt)
#    define HAVE_TDM 1
#  else
#    define HAVE_TDM 0
#  endif
#else
#  define HAVE_TDM 0
#endif

// D(16x16,f32) = A(16x4,f32) x B(4x16,f32) + C  — full-precision CDNA5 WMMA
__device__ __forceinline__ v8f wmma_f32_16x16x4(v2f a, v2f b, v8f c) {
  return __builtin_amdgcn_wmma_f32_16x16x4_f32(false, a, false, b, (short)0, c,
                                               false, false);
}

// Async global->LDS single-dword copy (per active lane), tracked by ASYNCcnt.
__device__ __forceinline__ void async_load_f32(const float* g, float* ldsDst) {
  // low 32 bits of a flat LDS pointer are the LDS byte offset
  unsigned int l = (unsigned int)(unsigned long long)ldsDst;
  unsigned long long ga = (unsigned long long)g;
  asm volatile("global_load_async_to_lds_b32 %0, %1, off"
               :: "v"(l), "v"(ga) : "memory");
}
__device__ __forceinline__ void wait_asynccnt0() {
  asm volatile("s_wait_asynccnt 0x0" ::: "memory");
}

#if HAVE_TDM
// TDM: DMA one 64x64 f32 tile (row-major, stride 64) from global into LDS.
__device__ __forceinline__ void tdm_load_64x64_f32(const float* gA, float* sA) {
  unsigned long long ga = (unsigned long long)gA;
  unsigned int lo = (unsigned int)(unsigned long long)sA;  // LDS byte offset
  // D# group0: count=1 | lds_addr | global_addr[56:0] | type=2
  uint32x4 g0 = { 1u, lo, (unsigned int)ga,
                  (unsigned int)((ga >> 32) & 0x1FFFFFFu) | 0x80000000u };
  // D# group1: data_size=4B; tensor_dim0/1=64; tile_dim0/1=64; dim0_stride=64
  int32x8 g1 = { (int)0x00020000,   // [17:16]=2 -> 4-byte elements
                 (int)0x00400000,   // tensor_dim0[15:0]=64 at bits[63:48]
                 (int)0x00400000,   // tensor_dim1[15:0]=64 at bits[95:80]
                 (int)0x00400000,   // tile_dim0=64 at bits[127:112]
                 64,                // tile_dim1=64 at bits[143:128]
                 64,                // tensor_dim0_stride[31:0]=64
                 0, 0 };
  int32x4 z4 = {};
#if __clang_major__ >= 23
  int32x8 z8 = {};
  __builtin_amdgcn_tensor_load_to_lds(g0, g1, z4, z4, z8, 0);
#else
  __builtin_amdgcn_tensor_load_to_lds(g0, g1, z4, z4, 0);
#endif
}
#endif

// Stage A (64x64 f32) into LDS: TDM when available, else per-thread loads.
__device__ __forceinline__ void load_A_to_lds(const float* __restrict__ Ag,
                                              float* sA, int tid) {
#if HAVE_TDM
  if (tid < 32) {                       // one wave issues the DMA
    tdm_load_64x64_f32(Ag, sA);
    __builtin_amdgcn_s_wait_tensorcnt(0);
  }
#else
  for (int j = 0; j < 16; ++j) sA[j * 256 + tid] = Ag[j * 256 + tid];
#endif
}

// ---------------------------------------------------------------------------
// 1) Gauss-Jordan inversion of A (64x64) + M = A^{-1} @ B_w  (single block)
// ---------------------------------------------------------------------------
__global__ __launch_bounds__(64) void prep_kernel(const float* __restrict__ Ag,
                                                  const float* __restrict__ Bw,
                                                  float* __restrict__ M) {
  __shared__ float aug[kN][2 * kN];
  const int t = threadIdx.x;
  for (int j = 0; j < kN; ++j) {
    aug[j][t]      = Ag[j * kN + t];
    aug[j][kN + t] = (j == t) ? 1.f : 0.f;
  }
  __syncthreads();
  for (int k = 0; k < kN; ++k) {
    const float ip = 1.0f / aug[k][k];
    __syncthreads();
    aug[k][t]      *= ip;
    aug[k][kN + t] *= ip;
    __syncthreads();
    if (t != k) {
      const float f = aug[t][k];
      for (int c = 0; c < 2 * kN; ++c) aug[t][c] -= f * aug[k][c];
    }
    __syncthreads();
  }
  for (int idx = t; idx < kN * kH; idx += kN) {
    const int n = idx / kH, h = idx % kH;
    float acc = 0.f;
    for (int k = 0; k < kN; ++k) acc += aug[n][kN + k] * Bw[k * kH + h];
    M[idx] = acc;
  }
}

// ---------------------------------------------------------------------------
// 2) V = X @ M^T   (4096x64, K=512) — one 16x16 tile per wave, WMMA f32
// ---------------------------------------------------------------------------
__global__ __launch_bounds__(256) void vproj_kernel(const float* __restrict__ X,
                                                    const float* __restrict__ M,
                                                    float* __restrict__ V) {
  const int lane = threadIdx.x & 31;
  const int wave = threadIdx.x >> 5;
  const int tile = blockIdx.x * 8 + wave;   // 256 x 4 tiles
  const int tm = tile >> 2;
  const int tn = tile & 3;
  const int ksel = (lane >> 4) << 1;
  const int arow = tm * 16 + (lane & 15);
  const int bcol = tn * 16 + (lane & 15);
  v8f acc = {};
  for (int k0 = 0; k0 < kH; k0 += 4) {
    v2f a, b;
    a.x = X[arow * kH + k0 + ksel];
    a.y = X[arow * kH + k0 + ksel + 1];
    b.x = M[bcol * kH + k0 + ksel];       // B[k][n] = M[n*H + k]
    b.y = M[bcol * kH + k0 + ksel + 1];
    acc = wmma_f32_16x16x4(a, b, acc);
  }
  const int rbase = tm * 16 + ((lane >> 4) << 3);
  for (int v = 0; v < 8; ++v) V[(rbase + v) * kN + bcol] = acc[v];
}

// ---------------------------------------------------------------------------
// 3) Per-chunk: A_bar_t = exp(dlt*A); u_t = A_bar_t v_t - v_t;
//    P = prod_t A_bar_t (WMMA f32 in LDS, register-staged), q = composite u
// ---------------------------------------------------------------------------
__global__ __launch_bounds__(256) void chunk_kernel(const float* __restrict__ Ag,
                                                    const float* __restrict__ delta,
                                                    const float* __restrict__ V,
                                                    float* __restrict__ U,
                                                    float* __restrict__ P,
                                                    float* __restrict__ Q) {
  __shared__ float sA[kN * kN];
  __shared__ float sAbar[kN * kN];
  __shared__ float sP[kN * kN];
  __shared__ float sq[2][kN];
  __shared__ float sv[kN];
  __shared__ float su[kN];
  const int tid = threadIdx.x;
  const int lane = tid & 31, wave = tid >> 5;
  const int bidx = blockIdx.x / kC, cidx = blockIdx.x % kC;
  const int t0 = cidx * kL;

  load_A_to_lds(Ag, sA, tid);                       // TDM DMA (wave 0)
  for (int j = 0; j < 16; ++j) {
    const int i = j * 256 + tid;
    sP[i] = ((i >> 6) == (i & 63)) ? 1.f : 0.f;     // identity
  }
  if (tid < kN) sq[0][tid] = 0.f;
  int cur = 0;
  __syncthreads();
  for (int tt = 0; tt < kL; ++tt) {
    const int s = t0 + tt;
    const float dlt = delta[bidx * kS + s];
    if (tid < kN) async_load_f32(&V[(bidx * kS + s) * kN + tid], &sv[tid]);
    for (int j = 0; j < 16; ++j)
      sAbar[j * 256 + tid] = __expf(dlt * sA[j * 256 + tid]);
    if (tid < kN) wait_asynccnt0();                 // v row landed in LDS
    __syncthreads();
    if (tid < kN) {                                 // u = A_bar v - v
      float acc = 0.f;
      for (int m = 0; m < kN; ++m) acc += sAbar[tid * kN + m] * sv[m];
      acc -= sv[tid];
      su[tid] = acc;
      U[(bidx * kS + s) * kN + tid] = acc;
    }
    __syncthreads();
    // P' = P @ A_bar : 16 tiles (4x4) of 16x16; 8 waves -> 2 tiles each,
    // results staged in registers so sP can be updated in place.
    const int ksel = (lane >> 4) << 1;
    v8f tacc[2];
    for (int ti = 0; ti < 2; ++ti) {
      const int tile = wave + ti * 8;
      const int tm = tile >> 2, tn = tile & 3;
      const int arow = tm * 16 + (lane & 15);
      const int bcol = tn * 16 + (lane & 15);
      v8f acc = {};
      for (int k0 = 0; k0 < kN; k0 += 4) {
        v2f a, b;
        a.x = sP[arow * kN + k0 + ksel];
        a.y = sP[arow * kN + k0 + ksel + 1];
        b.x = sAbar[(k0 + ksel) * kN + bcol];
        b.y = sAbar[(k0 + ksel + 1) * kN + bcol];
        acc = wmma_f32_16x16x4(a, b, acc);
      }
      tacc[ti] = acc;
    }
    if (tid < kN) {                                 // q' = q A_bar + u
      float acc = su[tid];
      for (int m = 0; m < kN; ++m) acc += sq[cur][m] * sAbar[m * kN + tid];
      sq[cur ^ 1][tid] = acc;
    }
    __syncthreads();                                // all reads of sP done
    for (int ti = 0; ti < 2; ++ti) {
      const int tile = wave + ti * 8;
      const int tm = tile >> 2, tn = tile & 3;
      const int bcol = tn * 16 + (lane & 15);
      const int rbase = tm * 16 + ((lane >> 4) << 3);
      for (int v = 0; v < 8; ++v) sP[(rbase + v) * kN + bcol] = tacc[ti][v];
    }
    __syncthreads();
    cur ^= 1;
  }
  float* Pout = P + blockIdx.x * kN * kN;
  for (int j = 0; j < 16; ++j) Pout[j * 256 + tid] = sP[j * 256 + tid];
  if (tid < kN) Q[blockIdx.x * kN + tid] = sq[cur][tid];
}

// ---------------------------------------------------------------------------
// 4) Serial combine across chunks: carry_c saved, h = h P_c + q_c
// ---------------------------------------------------------------------------
__global__ __launch_bounds__(64) void combine_kernel(const float* __restrict__ P,
                                                     const float* __restrict__ Q,
                                                     float* __restrict__ Carry) {
  __shared__ float sh[kN];
  const int t = threadIdx.x;
  const int b = blockIdx.x;
  sh[t] = 0.f;
  __syncthreads();
  for (int c = 0; c < kC; ++c) {
    const int idx = b * kC + c;
    Carry[idx * kN + t] = sh[t];
    const float* Pc = P + idx * kN * kN;
    float acc = Q[idx * kN + t];
    for (int m = 0; m < kN; ++m) acc += sh[m] * Pc[m * kN + t];
    __syncthreads();
    sh[t] = acc;
    __syncthreads();
  }
}

// ---------------------------------------------------------------------------
// 5) Per-chunk rescan with true carry: h_t = h_{t-1} A_bar_t + u_t
// ---------------------------------------------------------------------------
__global__ __launch_bounds__(256) void rescan_kernel(const float* __restrict__ Ag,
                                                     const float* __restrict__ delta,
                                                     const float* __restrict__ U,
                                                     const float* __restrict__ Carry,
                                                     float* __restrict__ HS) {
  __shared__ float sA[kN * kN];
  __shared__ float sAbar[kN * kN];
  __shared__ float sh[2][kN];
  __shared__ float sU[kN];
  const int tid = threadIdx.x;
  const int bidx = blockIdx.x / kC, cidx = blockIdx.x % kC;
  const int t0 = cidx * kL;
  load_A_to_lds(Ag, sA, tid);                       // TDM DMA (wave 0)
  if (tid < kN) sh[0][tid] = Carry[blockIdx.x * kN + tid];
  int cur = 0;
  __syncthreads();
  for (int tt = 0; tt < kL; ++tt) {
    const int s = t0 + tt;
    const float dlt = delta[bidx * kS + s];
    if (tid < kN) async_load_f32(&U[(bidx * kS + s) * kN + tid], &sU[tid]);
    for (int j = 0; j < 16; ++j)
      sAbar[j * 256 + tid] = __expf(dlt * sA[j * 256 + tid]);
    if (tid < kN) wait_asynccnt0();
    __syncthreads();
    if (tid < kN) {
      float acc = sU[tid];
      for (int m = 0; m < kN; ++m) acc += sh[cur][m] * sAbar[m * kN + tid];
      sh[cur ^ 1][tid] = acc;
      HS[(bidx * kS + s) * kN + tid] = acc;
    }
    __syncthreads();
    cur ^= 1;
  }
}

// ---------------------------------------------------------------------------
// 6) Y = HS @ Cw^T + X @ Dw^T + Cb + Db  — fused, one 16x16 tile per wave
// ---------------------------------------------------------------------------
__global__ __launch_bounds__(256) void output_kernel(const float* __restrict__ HS,
                                                     const float* __restrict__ X,
                                                     const float* __restrict__ Cw,
                                                     const float* __restrict__ Cb,
                                                     const float* __restrict__ Dw,
                                                     const float* __restrict__ Db,
                                                     float* __restrict__ Y) {
  const int lane = threadIdx.x & 31;
  const int wave = threadIdx.x >> 5;
  const int tile = blockIdx.x * 8 + wave;   // 256 row-tiles x 32 col-tiles
  const int tm = tile >> 5;
  const int tn = tile & 31;
  const int ksel = (lane >> 4) << 1;
  const int arow = tm * 16 + (lane & 15);
  const int bcol = tn * 16 + (lane & 15);
  v8f acc = {};
  for (int k0 = 0; k0 < kN; k0 += 4) {      // HS @ Cw^T  (B[k][h] = Cw[h*N+k])
    v2f a, b;
    a.x = HS[arow * kN + k0 + ksel];
    a.y = HS[arow * kN + k0 + ksel + 1];
    b.x = Cw[bcol * kN + k0 + ksel];
    b.y = Cw[bcol * kN + k0 + ksel + 1];
    acc = wmma_f32_16x16x4(a, b, acc);
  }
  for (int k0 = 0; k0 < kH; k0 += 4) {      // X @ Dw^T   (B[k][g] = Dw[g*H+k])
    __builtin_prefetch(&X[arow * kH + k0 + 64], 0, 1);   // global_prefetch
    v2f a, b;
    a.x = X[arow * kH + k0 + ksel];
    a.y = X[arow * kH + k0 + ksel + 1];
    b.x = Dw[bcol * kH + k0 + ksel];
    b.y = Dw[bcol * kH + k0 + ksel + 1];
    acc = wmma_f32_16x16x4(a, b, acc);
  }
  const float bias = Cb[bcol] + Db[bcol];
  const int rbase = tm * 16 + ((lane >> 4) << 3);
  for (int v = 0; v < 8; ++v) Y[(rbase + v) * kH + bcol] = acc[v] + bias;
}

extern "C" void kernel_launch(void* const* d_in, const int* in_sizes, int n_in,
                              void* d_out, int out_size, void* d_ws, size_t ws_size,
                              hipStream_t stream) {
  const float* X     = (const float*)d_in[0];
  const float* delta = (const float*)d_in[1];
  const float* A     = (const float*)d_in[2];
  const float* Bw    = (const float*)d_in[3];
  const float* Cw    = (const float*)d_in[4];
  const float* Cb    = (const float*)d_in[5];
  const float* Dw    = (const float*)d_in[6];
  const float* Db    = (const float*)d_in[7];
  float* Y = (float*)d_out;

  float* ws    = (float*)d_ws;
  float* M     = ws;                        // N*H         = 32768
  float* V     = M + kN * kH;               // R*N         = 262144
  float* U     = V + kR * kN;               // R*N
  float* P     = U + kR * kN;               // B*C*N*N     = 262144
  float* Q     = P + kB * kC * kN * kN;     // B*C*N
  float* Carry = Q + kB * kC * kN;          // B*C*N
  float* HS    = Carry + kB * kC * kN;      // R*N
  (void)in_sizes; (void)n_in; (void)out_size; (void)ws_size;

  prep_kernel<<<1, 64, 0, stream>>>(A, Bw, M);
  vproj_kernel<<<(kR / 16) * (kN / 16) / 8, 256, 0, stream>>>(X, M, V);
  chunk_kernel<<<kB * kC, 256, 0, stream>>>(A, delta, V, U, P, Q);
  combine_kernel<<<kB, 64, 0, stream>>>(P, Q, Carry);
  rescan_kernel<<<kB * kC, 256, 0, stream>>>(A, delta, U, Carry, HS);
  output_kernel<<<(kR / 16) * (kH / 16) / 8, 256, 0, stream>>>(HS, X, Cw, Cb, Dw, Db, Y);
}